// PairWiseRankingLoss_3332894622720
// MI455X (gfx1250) — compile-verified
//
#include <hip/hip_runtime.h>

// PairWiseRankingLoss, gama=1, B=128, L=1024.
// Key identity: sigmoid outputs p in [0,1] => gama + p_i - p_j >= 0 always,
// so relu never clips and the O(B*L^2) pair sum collapses exactly to
//   loss_b = n0*n1*gama + n1*S0 - n0*S1
// with S0 = sum p over y==0, S1 = sum p over y==1, n0 = L - n1.
// Row sums are computed with V_WMMA_F32_16X16X4_F32 (full f32 WMMA):
// A = 16x4 slice of masked values, B = ones(4x16), accumulated over K.

typedef float v2f __attribute__((ext_vector_type(2)));
typedef float v8f __attribute__((ext_vector_type(8)));

#define PRL_GAMA 1.0f
#define PRL_B    128
#define PRL_L    1024

// Native-precision sigmoid: v_exp_f32 + v_rcp_f32 (no IEEE div sequence).
__device__ __forceinline__ float fast_sigmoid(float x) {
    return __builtin_amdgcn_rcpf(1.0f + __expf(-x));
}

// One wave (32 threads) per block; block b handles rows 16b .. 16b+15.
// A-matrix VGPR layout for 16x4 f32 (ISA 7.12.2):
//   lanes 0-15 : (M=lane,    K=0,1)  -> float2 at row offset k+0
//   lanes 16-31: (M=lane-16, K=2,3)  -> float2 at row offset k+2
// C/D layout: acc[r], lanes 0-15 -> row r ; lanes 16-31 -> row r+8
// (row sum replicated across all 16 N columns since B is all-ones).
__global__ __launch_bounds__(32) void prl_rowsum_wmma_kernel(
    const float* __restrict__ logit,
    const int*   __restrict__ label,
    float*       __restrict__ partial)
{
    const int lane = threadIdx.x;                 // 0..31, full wave
    const int row  = blockIdx.x * 16 + (lane & 15);
    const int koff = (lane >> 4) * 2;             // 0 or 2

    const float* prow = logit + row * PRL_L;
    const int*   yrow = label + row * PRL_L;

    v8f accSA = {};   // sum p            (all elements)
    v8f accS1 = {};   // sum p * [y==1]
    v8f accN1 = {};   // sum     [y==1]
    const v2f bOnes = {1.0f, 1.0f};               // B = ones(4x16)

    for (int k = 0; k < PRL_L; k += 4) {
        const int idx = k + koff;
        v2f  x  = *(const v2f*)(prow + idx);      // 8B aligned (idx even)
        int2 yy = *(const int2*)(yrow + idx);

        float p0 = fast_sigmoid(x.x);
        float p1 = fast_sigmoid(x.y);
        float f0 = (float)yy.x;                   // labels are {0,1}
        float f1 = (float)yy.y;

        v2f aSA = {p0,      p1     };
        v2f aS1 = {p0 * f0, p1 * f1};
        v2f aN1 = {f0,      f1     };

        accSA = __builtin_amdgcn_wmma_f32_16x16x4_f32(
                    false, aSA, false, bOnes, (short)0, accSA, false, false);
        accS1 = __builtin_amdgcn_wmma_f32_16x16x4_f32(
                    false, aS1, false, bOnes, (short)0, accS1, false, false);
        accN1 = __builtin_amdgcn_wmma_f32_16x16x4_f32(
                    false, aN1, false, bOnes, (short)0, accN1, false, false);
    }

    // Extract 16 row sums and fold into the closed-form per-row loss.
    float total = 0.0f;
#pragma unroll
    for (int r = 0; r < 8; ++r) {
        {   // row r  -> lanes 0-15 of acc component r (use lane 0)
            float sa = __shfl(accSA[r], 0, 32);
            float s1 = __shfl(accS1[r], 0, 32);
            float n1 = __shfl(accN1[r], 0, 32);
            float n0 = (float)PRL_L - n1;
            total += n0 * n1 * PRL_GAMA + n1 * (sa - s1) - n0 * s1;
        }
        {   // row r+8 -> lanes 16-31 of acc component r (use lane 16)
            float sa = __shfl(accSA[r], 16, 32);
            float s1 = __shfl(accS1[r], 16, 32);
            float n1 = __shfl(accN1[r], 16, 32);
            float n0 = (float)PRL_L - n1;
            total += n0 * n1 * PRL_GAMA + n1 * (sa - s1) - n0 * s1;
        }
    }

    if (lane == 0) partial[blockIdx.x] = total;
}

// Tiny finalize: sum the 8 per-block partials into the scalar output.
// (Separate kernel => no atomics, no need to pre-zero d_out, graph-safe.)
__global__ __launch_bounds__(32) void prl_finalize_kernel(
    const float* __restrict__ partial, float* __restrict__ out)
{
    if (threadIdx.x == 0 && blockIdx.x == 0) {
        float s = 0.0f;
#pragma unroll
        for (int i = 0; i < PRL_B / 16; ++i) s += partial[i];
        out[0] = s;
    }
}

extern "C" void kernel_launch(void* const* d_in, const int* in_sizes, int n_in,
                              void* d_out, int out_size, void* d_ws, size_t ws_size,
                              hipStream_t stream) {
    const float* logit = (const float*)d_in[0];   // Y_logit [128,1024] f32
    const int*   label = (const int*)  d_in[1];   // Y       [128,1024] i32
    float* partial = (float*)d_ws;                // 8 floats of scratch
    float* out     = (float*)d_out;               // scalar f32

    prl_rowsum_wmma_kernel<<<PRL_B / 16, 32, 0, stream>>>(logit, label, partial);
    prl_finalize_kernel<<<1, 32, 0, stream>>>(partial, out);
}